// GCN_26147760898239
// MI455X (gfx1250) — compile-verified
//
#include <hip/hip_runtime.h>
#include <hip/hip_bf16.h>

typedef __attribute__((ext_vector_type(2))) float v2f;
typedef __attribute__((ext_vector_type(8))) float v8f;

#define Ntot 32
#define Cch  64
#define Tlen 512
#define Vj   25
#define Hh   3
#define TB   8            // t-values per workgroup
#define UP   32           // u/v padded to 32 (zero-padded -> exact math)
#define COLS (TB*UP)      // 256 stage-A columns
#define BN_EPS 1e-5f

#define GLOBAL_AS __attribute__((address_space(1)))
#define LDS_AS    __attribute__((address_space(3)))

#if defined(__has_builtin)
#if __has_builtin(__builtin_amdgcn_global_load_async_to_lds_b32)
#define HAVE_ASYNC_LDS 1
#endif
#endif

// ---------------------------------------------------------------------------
// K0: zero the per-channel stats accumulator (sum, sumsq) in workspace.
// ---------------------------------------------------------------------------
__global__ void gcn_zero_stats(float* __restrict__ s) {
    s[threadIdx.x] = 0.0f;   // launched <<<1,128>>>
}

// ---------------------------------------------------------------------------
// K1: fused  out_pre = sum_h (W_h * X) * A_h^T + bias   via f32 WMMA,
//     plus per-channel sum/sumsq accumulation for BatchNorm.
// Grid: (T/TB, N), Block: 256 threads = 8 waves (wave32).
// ---------------------------------------------------------------------------
__global__ __launch_bounds__(256)
void gcn_fused_wmma(const float* __restrict__ xg,   // [N,C,T,V]
                    const float* __restrict__ Ag,   // [H,V,V]
                    const float* __restrict__ Wg,   // [H,C,C]
                    const float* __restrict__ bg,   // [H,C]
                    float* __restrict__ outPre,     // [N,C,T,V] pre-BN
                    float* __restrict__ gstats)     // [2*C]
{
    // xs: channel-pair interleaved: element (c,col) at (c>>1)*512 + col*2 + (c&1)
    // -> X[k][col],X[k+1][col] (k even) are one aligned 8B ds_load_b64.
    __shared__ __align__(16) float xs [Cch*COLS];   // 64KB
    __shared__ __align__(16) float ys [Cch*COLS];   // 64KB : Y[row=o*8+t][u]
    __shared__ __align__(16) float whs[Cch*Cch];    // 16KB : W_h[o][c]
    __shared__ __align__(16) float ash[Hh*UP*UP];   // 12KB : A[h][v][u] padded
    __shared__ float bsum_s [Cch];
    __shared__ float stats_s[2*Cch];

    const int tid    = threadIdx.x;
    const int wave   = tid >> 5;
    const int lane   = tid & 31;
    const int lane16 = lane & 15;
    const int khalf  = (lane & 16) ? 2 : 0;   // ISA A/B fragment K-split
    const int t0     = blockIdx.x * TB;
    const int n      = blockIdx.y;

    // ---- load X tile (zero-pad u to 32), async to LDS when available ----
    const size_t xbase = (size_t)n * Cch * (Tlen * Vj);
    for (int idx = tid; idx < Cch*COLS; idx += 256) {
        int c    = idx >> 8;          // / COLS
        int col  = idx & (COLS-1);
        int t    = col >> 5;
        int u    = col & 31;
        int lidx = (c >> 1)*(2*COLS) + col*2 + (c & 1);   // interleaved
        if (u < Vj) {
            size_t off = xbase + (size_t)c*(Tlen*Vj) + (size_t)(t0 + t)*Vj + u;
#if HAVE_ASYNC_LDS
            __builtin_amdgcn_global_load_async_to_lds_b32(
                (GLOBAL_AS int*)(xg + off),
                (LDS_AS int*)(&xs[lidx]), 0, 0);
#else
            xs[lidx] = xg[off];
#endif
            if (t0 + TB < Tlen)
                __builtin_prefetch(&xg[off + TB*Vj], 0, 3);  // global_prefetch
        } else {
            xs[lidx] = 0.0f;
        }
    }
    // ---- load A for all heads, zero-padded to 32x32 ----
    for (int idx = tid; idx < Hh*UP*UP; idx += 256) {
        int h = idx >> 10;
        int v = (idx >> 5) & 31;
        int u = idx & 31;
        ash[idx] = (v < Vj && u < Vj) ? Ag[h*(Vj*Vj) + v*Vj + u] : 0.0f;
    }
    if (tid < Cch)   bsum_s[tid]  = bg[tid] + bg[Cch + tid] + bg[2*Cch + tid];
    if (tid < 2*Cch) stats_s[tid] = 0.0f;
#if HAVE_ASYNC_LDS
#if __has_builtin(__builtin_amdgcn_s_wait_asynccnt)
    __builtin_amdgcn_s_wait_asynccnt(0);
#else
    asm volatile("s_wait_asynccnt 0" ::: "memory");
#endif
#endif
    __syncthreads();

    v8f accO[8] = {};   // persistent OUT accumulators across heads

    for (int h = 0; h < Hh; ++h) {
        // ---- load W_h ----
        for (int idx = tid; idx < Cch*Cch; idx += 256)
            whs[idx] = Wg[h*Cch*Cch + idx];
        __syncthreads();

        // ---- Stage A: Y = W_h (64x64) * X (64 x 256), K=c ----
        // 4 M-tiles x 16 col-tiles = 64 tiles; 8 per wave.
        #pragma unroll
        for (int i = 0; i < 8; ++i) {
            int tile  = wave*8 + i;
            int mtile = tile >> 4;          // o-tile 0..3
            int ctile = tile & 15;          // col-tile 0..15
            int am = mtile*16 + lane16;     // output row o
            int bj = ctile*16 + lane16;     // column (t,u)
            v8f acc = {};
            #pragma unroll
            for (int kc = 0; kc < 16; ++kc) {
                int ka = kc*4 + khalf;      // channel c (even)
                v2f af = *(const v2f*)&whs[am*Cch + ka];        // W_h[o][c..c+1]
                v2f bf = *(const v2f*)&xs[(ka >> 1)*(2*COLS) + bj*2]; // X[c..c+1][col]
                acc = __builtin_amdgcn_wmma_f32_16x16x4_f32(
                        false, af, false, bf, (short)0, acc, false, false);
            }
            int baser = mtile*16 + ((lane & 16) ? 8 : 0);  // C/D VGPR layout
            #pragma unroll
            for (int r = 0; r < 8; ++r)
                ys[(baser + r)*COLS + bj] = acc[r];
        }
        __syncthreads();

        // ---- Stage B: OUT[(o,t), v] += Y[(o,t), u] * A_h[v][u], K=u(32) ----
        // rows = o*8+t (512) -> 32 M-tiles x 2 v-tiles = 64 tiles; 8 per wave.
        #pragma unroll
        for (int i = 0; i < 8; ++i) {
            int tile  = wave*8 + i;
            int mtile = tile >> 1;          // row-tile 0..31
            int vtile = tile & 1;
            int row  = mtile*16 + lane16;   // row = o*8 + t -> ys[row*32 + u]
            int vcol = vtile*16 + lane16;
            #pragma unroll
            for (int kc = 0; kc < 8; ++kc) {
                int ka = kc*4 + khalf;      // u (even)
                v2f af = *(const v2f*)&ys[row*UP + ka];
                v2f bf = *(const v2f*)&ash[h*(UP*UP) + vcol*UP + ka]; // A_h[v][u..u+1]
                accO[i] = __builtin_amdgcn_wmma_f32_16x16x4_f32(
                        false, af, false, bf, (short)0, accO[i], false, false);
            }
        }
        __syncthreads();
    }

    // ---- epilogue: +bias, store pre-BN, accumulate BN stats ----
    #pragma unroll
    for (int i = 0; i < 8; ++i) {
        int tile  = wave*8 + i;
        int mtile = tile >> 1;
        int vtile = tile & 1;
        int v = vtile*16 + lane16;
        if (v >= Vj) continue;                       // padded columns dropped
        int rbase = mtile*16 + ((lane & 16) ? 8 : 0);
        int o = rbase >> 3;                          // same o for all 8 rows
        float bias = bsum_s[o];
        float s1 = 0.0f, s2 = 0.0f;
        size_t obase = (((size_t)n*Cch + o)*Tlen + t0)*Vj + v;  // t = r
        #pragma unroll
        for (int r = 0; r < 8; ++r) {
            float val = accO[i][r] + bias;
            outPre[obase + (size_t)r*Vj] = val;
            s1 += val; s2 += val*val;
        }
        atomicAdd(&stats_s[o],        s1);
        atomicAdd(&stats_s[Cch + o],  s2);
    }
    __syncthreads();
    if (tid < 2*Cch) atomicAdd(&gstats[tid], stats_s[tid]);
}

// ---------------------------------------------------------------------------
// K2: finalize BN stats, apply scale/shift + residual + ReLU in place.
// Grid: N*C*T*V/256 blocks; channel is uniform per block (T*V = 50*256).
// ---------------------------------------------------------------------------
__global__ __launch_bounds__(256)
void gcn_bn_res_relu(const float* __restrict__ xg,
                     const float* __restrict__ gamma,
                     const float* __restrict__ beta,
                     const float* __restrict__ gstats,
                     float* __restrict__ out)
{
    __shared__ float sc[Cch], sh[Cch];
    const int tid = threadIdx.x;
    if (tid < Cch) {
        const float invcnt = 1.0f / (float)(Ntot * Tlen * Vj);
        float mean = gstats[tid] * invcnt;
        float var  = gstats[Cch + tid] * invcnt - mean*mean;  // biased var
        float inv  = rsqrtf(var + BN_EPS);
        float g    = gamma[tid];
        sc[tid] = g * inv;
        sh[tid] = beta[tid] - g * mean * inv;
    }
    __syncthreads();
    const int c = (blockIdx.x / 50) & 63;     // T*V = 12800 = 50 blocks/channel
    size_t idx = (size_t)blockIdx.x * 256 + tid;
    float pre = out[idx];
    float val = sc[c] * pre + sh[c] + xg[idx];
    out[idx] = fmaxf(val, 0.0f);
}

// ---------------------------------------------------------------------------
extern "C" void kernel_launch(void* const* d_in, const int* in_sizes, int n_in,
                              void* d_out, int out_size, void* d_ws, size_t ws_size,
                              hipStream_t stream) {
    (void)in_sizes; (void)n_in; (void)out_size; (void)ws_size;
    const float* x     = (const float*)d_in[0];
    const float* A     = (const float*)d_in[1];
    const float* W     = (const float*)d_in[2];
    const float* b     = (const float*)d_in[3];
    const float* gamma = (const float*)d_in[4];
    const float* beta  = (const float*)d_in[5];
    float* out   = (float*)d_out;            // doubles as pre-BN buffer
    float* stats = (float*)d_ws;             // 128 floats

    gcn_zero_stats<<<1, 128, 0, stream>>>(stats);
    gcn_fused_wmma<<<dim3(Tlen/TB, Ntot), 256, 0, stream>>>(x, A, W, b, out, stats);
    gcn_bn_res_relu<<<dim3((Ntot*Cch*Tlen*Vj)/256), 256, 0, stream>>>(
        x, gamma, beta, stats, out);
}